// ElectrostaticCorrection_1821066133972
// MI455X (gfx1250) — compile-verified
//
#include <hip/hip_runtime.h>
#include <hip/hip_bf16.h>
#include <math.h>

// ---------------------------------------------------------------------------
// ElectrostaticCorrection for MI455X (gfx1250, wave32)
//
// Phase 1: charges = silu(h @ W1 + b1) @ W2 + b2      -> V_WMMA_F32_16X16X4_F32
//          A-tile staged to LDS via GLOBAL_LOAD_ASYNC_TO_LDS_B128 (ASYNCcnt)
// Phase 2: per-graph mean subtraction (fused into phase 3)
// Phase 3: block-structured pairwise erfc energy (batch is sorted -> each
//          graph is a contiguous run; O(sum cnt_g^2) ~= 0.8M pairs instead of
//          the reference's dense 6144^2 = 37.7M).
// ---------------------------------------------------------------------------

typedef __attribute__((ext_vector_type(2))) float v2f;
typedef __attribute__((ext_vector_type(8))) float v8f;

#define DIM   256      // scalar_dim
#define HID   128      // DIM/2
#define TM    16       // WMMA tile M
#define LDA   260      // LDS row stride for A tile (260 % 64 == 4 -> conflict-free)

// -------------------------------------------------------------------------
// Kernel 0: graph start offsets via binary search (batch is sorted).
// start[g] = first i with batch[i] >= g ; start[G] = N.
// -------------------------------------------------------------------------
__global__ void ec_offsets_kernel(const int* __restrict__ batch,
                                  int* __restrict__ start, int N, int G) {
    int g = blockIdx.x * blockDim.x + threadIdx.x;
    if (g > G) return;
    int lo = 0, hi = N;
    while (lo < hi) {
        int mid = (lo + hi) >> 1;
        if (batch[mid] < g) lo = mid + 1; else hi = mid;
    }
    start[g] = lo;
}

// -------------------------------------------------------------------------
// Kernel 1: charge MLP with fp32 WMMA.
// Grid: N/16 blocks, 256 threads (8 wave32). Each block owns 16 rows of h;
// wave w owns columns [16w, 16w+16) of the 128-wide hidden layer.
//
// V_WMMA_F32_16X16X4_F32 operand layout (ISA 7.12.2, 32-bit matrices):
//   A (16x4, MxK): lanes 0-15: M=lane, v0=K0, v1=K1 ; lanes 16-31: M=lane-16,
//                  v0=K2, v1=K3.
//   B (4x16, KxN): rows striped across lanes: lanes 0-15: N=lane, v0=K0,
//                  v1=K1 ; lanes 16-31: N=lane-16, v0=K2, v1=K3.
//   C/D (16x16):   VGPR r, lanes 0-15 -> (M=r, N=lane); lanes 16-31 -> M=r+8.
// -------------------------------------------------------------------------
__global__ __launch_bounds__(256)
void ec_mlp_wmma_kernel(const float* __restrict__ h,
                        const float* __restrict__ W1,
                        const float* __restrict__ b1,
                        const float* __restrict__ W2,
                        const float* __restrict__ b2,
                        float* __restrict__ raw_q) {
    __shared__ float Ah[TM * LDA];          // 16 x 256 A tile, padded stride
    __shared__ float P[8][TM];              // per-wave row partials

    const int tid  = threadIdx.x;
    const int row0 = blockIdx.x * TM;

    // Async copy of the 16x256 A tile straight into LDS (no VGPR bounce).
    // 16 rows x 1KB; chunk c covers row (c>>6), floats [(c&63)*4, +4).
    // 64 consecutive chunks = one fully coalesced 1KB row segment.
    const unsigned lds_base = (unsigned)(uintptr_t)&Ah[0];
#pragma unroll
    for (int k = 0; k < 4; ++k) {
        const int c   = tid + (k << 8);
        const int r   = c >> 6;
        const int col = (c & 63) << 2;
        const unsigned     lds_off = lds_base + ((unsigned)(r * LDA + col) << 2);
        const float* gp = h + (size_t)(row0 + r) * DIM + col;
        asm volatile("global_load_async_to_lds_b128 %0, %1, off"
                     :: "v"(lds_off), "v"(gp) : "memory");
    }
    asm volatile("s_wait_asynccnt 0" ::: "memory");
    __syncthreads();

    const int wave  = tid >> 5;             // 0..7
    const int lane  = tid & 31;
    const int half  = lane >> 4;            // 0: K{0,1} / rows 0..7 ; 1: K{2,3} / rows 8..15
    const int l16   = lane & 15;
    const int mycol = wave * 16 + l16;      // hidden column this lane owns in B/C

    v8f acc = {};
    for (int k0 = 0; k0 < DIM; k0 += 4) {
        const int ka = k0 + 2 * half;
        v2f a, b;
        a[0] = Ah[l16 * LDA + ka];
        a[1] = Ah[l16 * LDA + ka + 1];
        b[0] = W1[(size_t)ka       * HID + mycol];
        b[1] = W1[(size_t)(ka + 1) * HID + mycol];
        // 8 args: (neg_a, A, neg_b, B, c_mod, C, reuse_a, reuse_b)
        acc = __builtin_amdgcn_wmma_f32_16x16x4_f32(
            false, a, false, b, (short)0, acc, false, false);
    }

    // Fused epilogue: bias + SiLU + multiply by W2 column weight.
    // v_rcp_f32 (~2^-22 rel err) instead of the full IEEE divide sequence.
    const float b1v = b1[mycol];
    const float w2v = W2[mycol];
    float part[8];
#pragma unroll
    for (int r = 0; r < 8; ++r) {
        const float x = acc[r] + b1v;
        const float s = x * __builtin_amdgcn_rcpf(1.0f + __expf(-x));  // SiLU
        part[r] = s * w2v;
    }
    // Reduce across the 16 lanes of each half (columns of this wave's tile).
#pragma unroll
    for (int r = 0; r < 8; ++r) {
        float v = part[r];
#pragma unroll
        for (int off = 1; off < 16; off <<= 1)
            v += __shfl_xor(v, off, 32);           // xor<16 stays within half
        part[r] = v;
    }
    if (l16 == 0) {
#pragma unroll
        for (int r = 0; r < 8; ++r)
            P[wave][r + 8 * half] = part[r];       // row m = r + 8*half
    }
    __syncthreads();

    // Cross-wave reduction over the 8 column tiles -> raw charge per atom.
    if (tid < TM) {
        float q = b2[0];
#pragma unroll
        for (int w = 0; w < 8; ++w) q += P[w][tid];
        raw_q[row0 + tid] = q;
    }
}

// -------------------------------------------------------------------------
// Kernel 2: one workgroup per graph.
//  a) reduce raw charges -> mean (charge neutrality)
//  b) write centered charges to out[G + i]
//  c) LDS-tiled pairwise erfc energy -> out[g]
// -------------------------------------------------------------------------
__global__ __launch_bounds__(256)
void ec_pair_kernel(const float* __restrict__ pos,
                    const int* __restrict__ start,
                    const float* __restrict__ raw_q,
                    const float* __restrict__ log_sigma,
                    float* __restrict__ out, int G) {
    const int g = blockIdx.x;
    const int t = threadIdx.x;
    const int s = start[g];
    const int e = start[g + 1];
    const int cnt = e - s;

    __shared__ float red[256];
    __shared__ float jx[256], jy[256], jz[256], jq[256];

    // ---- per-graph mean of raw charges ----
    float sum = 0.0f;
    for (int i = s + t; i < e; i += 256) sum += raw_q[i];
    red[t] = sum;
    __syncthreads();
    for (int off = 128; off > 0; off >>= 1) {
        if (t < off) red[t] += red[t + off];
        __syncthreads();
    }
    const float mean = red[0] * __builtin_amdgcn_rcpf(fmaxf((float)cnt, 1.0f));
    __syncthreads();

    const float sigma     = log1pf(__expf(log_sigma[0])) + 1e-4f;  // softplus + eps
    const float inv_sigma = __builtin_amdgcn_rcpf(sigma);

    // ---- tiled pairwise loop ----
    float etot = 0.0f;
    const int rounds = (cnt + 255) >> 8;      // i-atoms per thread, uniform
    for (int r = 0; r < rounds; ++r) {
        const int  i    = s + (r << 8) + t;
        const bool live = (i < e);
        float qi = 0.f, pix = 0.f, piy = 0.f, piz = 0.f;
        if (live) {
            qi  = raw_q[i] - mean;
            pix = pos[3 * i];
            piy = pos[3 * i + 1];
            piz = pos[3 * i + 2];
            out[G + i] = qi;                  // centered charges output
        }
        float acc = 0.0f;
        for (int c0 = 0; c0 < cnt; c0 += 256) {         // uniform chunk loop
            const int j = s + c0 + t;
            if (j < e) {
                jx[t] = pos[3 * j];
                jy[t] = pos[3 * j + 1];
                jz[t] = pos[3 * j + 2];
                jq[t] = raw_q[j] - mean;
            }
            __syncthreads();
            const int m = min(256, cnt - c0);
            if (live) {
                for (int jj = 0; jj < m; ++jj) {
                    const int gj = s + c0 + jj;
                    if (gj == i) continue;              // skip self
                    const float dx = jx[jj] - pix;
                    const float dy = jy[jj] - piy;
                    const float dz = jz[jj] - piz;
                    const float d  =
                        __builtin_amdgcn_sqrtf(dx * dx + dy * dy + dz * dz);
                    acc += jq[jj] * erfcf(d * inv_sigma)
                                  * __builtin_amdgcn_rcpf(d + 1e-8f);
                }
            }
            __syncthreads();
        }
        etot += 0.5f * qi * acc;
    }

    // ---- block reduction -> e_elec[g] ----
    red[t] = etot;
    __syncthreads();
    for (int off = 128; off > 0; off >>= 1) {
        if (t < off) red[t] += red[t + off];
        __syncthreads();
    }
    if (t == 0) out[g] = red[0];
}

// -------------------------------------------------------------------------
// Launch. Inputs (setup_inputs order):
//   0: h_scalar [N,256] f32   1: positions [N,3] f32   2: W1 [256,128] f32
//   3: b1 [128] f32           4: W2 [128] f32          5: b2 [1] f32
//   6: log_sigma [1] f32      7: batch [N] i32
// Output: e_elec [G] ++ charges [N]  (f32), G = out_size - N.
// -------------------------------------------------------------------------
extern "C" void kernel_launch(void* const* d_in, const int* in_sizes, int n_in,
                              void* d_out, int out_size, void* d_ws, size_t ws_size,
                              hipStream_t stream) {
    const float* h   = (const float*)d_in[0];
    const float* pos = (const float*)d_in[1];
    const float* W1  = (const float*)d_in[2];
    const float* b1  = (const float*)d_in[3];
    const float* W2  = (const float*)d_in[4];
    const float* b2  = (const float*)d_in[5];
    const float* ls  = (const float*)d_in[6];
    const int*   bat = (const int*)d_in[7];

    const int N = in_sizes[7];
    const int G = out_size - N;

    int*   start = (int*)d_ws;                      // G+1 ints
    float* raw_q = (float*)((char*)d_ws + 1024);    // N floats, 1KB-aligned

    ec_offsets_kernel<<<1, 64, 0, stream>>>(bat, start, N, G);
    ec_mlp_wmma_kernel<<<N / TM, 256, 0, stream>>>(h, W1, b1, W2, b2, raw_q);
    ec_pair_kernel<<<G, 256, 0, stream>>>(pos, start, raw_q, ls, (float*)d_out, G);
}